// MedianFilter_3092376453297
// MI455X (gfx1250) — compile-verified
//
#include <hip/hip_runtime.h>
#include <hip/hip_bf16.h>
#include <stdint.h>

// 3x3 median filter, zero padding, B=128, C=1, H=W=512, fp32.
// Bandwidth-bound (>=268 MB @ 23.3 TB/s ~ 11.5us floor). Strategy:
//  - stage a (TW+2)x(TH+2) halo tile into LDS with CDNA5 async global->LDS
//    loads (ASYNCcnt), zero-fill the out-of-image halo with DS stores,
//  - s_wait_asynccnt 0 + barrier,
//  - each thread reads a 6-wide x 3-tall strip from LDS (b128+b64 DS loads),
//    reduces the 6 column triples with v_min3/v_med3/v_max3, combines them
//    with the exact identity med9 = med3(max(lows), med3(mids), min(highs)),
//  - one coalesced global_store_b128 (4 output pixels) per thread.

#define TW 128          // tile width  (output cols per block)
#define TH 8            // tile height (output rows per block)
#define LW (TW + 2)     // 130 loaded cols
#define LH (TH + 2)     // 10 loaded rows
#define LSTRIDE 132     // padded LDS row stride (floats); 132*4B keeps 16B align
#define NTHREADS 256

typedef float f4 __attribute__((ext_vector_type(4)));
typedef float f2 __attribute__((ext_vector_type(2)));

__device__ __forceinline__ void async_load_dword(uint32_t lds_byte_off,
                                                 const float* gaddr) {
  // CDNA5 async copy: memory -> LDS, no VGPR staging, tracked by ASYNCcnt.
  asm volatile("global_load_async_to_lds_b32 %0, %1, off"
               :
               : "v"(lds_byte_off), "v"(gaddr)
               : "memory");
}

__device__ __forceinline__ float med3(float a, float b, float c) {
  return __builtin_amdgcn_fmed3f(a, b, c);   // v_med3 (1 VALU op)
}
__device__ __forceinline__ float min3(float a, float b, float c) {
  return fminf(fminf(a, b), c);              // fuses to v_min3
}
__device__ __forceinline__ float max3(float a, float b, float c) {
  return fmaxf(fmaxf(a, b), c);              // fuses to v_max3
}

__global__ __launch_bounds__(NTHREADS)
void median3x3_kernel(const float* __restrict__ x, float* __restrict__ out,
                      int H, int W) {
  __shared__ float tile[LH * LSTRIDE];

  const int tid   = threadIdx.x;
  const int tileX = blockIdx.x * TW;
  const int tileY = blockIdx.y * TH;
  const int img   = blockIdx.z;

  const float* src = x   + (size_t)img * H * W;
  float*       dst = out + (size_t)img * H * W;

  // Low 32 bits of a flat shared-memory pointer == LDS byte offset (ISA 10.2).
  const uint32_t lds_base = (uint32_t)(uintptr_t)(&tile[0]);

  // ---- Stage halo tile (LH x LW) into LDS ----
  for (int idx = tid; idx < LH * LW; idx += NTHREADS) {
    const int ly = idx / LW;
    const int lx = idx - ly * LW;
    const int gy = tileY - 1 + ly;
    const int gx = tileX - 1 + lx;
    const int loff_elems = ly * LSTRIDE + lx;
    if ((unsigned)gy < (unsigned)H && (unsigned)gx < (unsigned)W) {
      async_load_dword(lds_base + (uint32_t)loff_elems * 4u,
                       src + (size_t)gy * W + gx);
    } else {
      tile[loff_elems] = 0.0f;  // zero padding, normal DS store (DScnt)
    }
  }

  asm volatile("s_wait_asynccnt 0x0" ::: "memory");
  __syncthreads();

  // ---- Compute: each thread -> 4 horizontally adjacent output pixels ----
  const int q  = tid & 31;   // column-quad index: output cols 4q..4q+3
  const int oy = tid >> 5;   // output row within tile: 0..7

  // Window for output col ox spans LDS cols ox..ox+2, so the quad needs
  // LDS cols 4q..4q+5 over rows oy..oy+2. Base (oy+r)*132 + 4q is 16B-aligned.
  float v[3][6];
#pragma unroll
  for (int r = 0; r < 3; ++r) {
    const float* rowp = &tile[(oy + r) * LSTRIDE + 4 * q];
    f4 a = *(const f4*)(rowp);       // ds_load_b128
    f2 b = *(const f2*)(rowp + 4);   // ds_load_b64
    v[r][0] = a.x; v[r][1] = a.y; v[r][2] = a.z; v[r][3] = a.w;
    v[r][4] = b.x; v[r][5] = b.y;
  }

  // Reduce each of the 6 column triples once (shared across the 4 windows).
  float lo[6], mi[6], hi[6];
#pragma unroll
  for (int j = 0; j < 6; ++j) {
    lo[j] = min3(v[0][j], v[1][j], v[2][j]);
    mi[j] = med3(v[0][j], v[1][j], v[2][j]);
    hi[j] = max3(v[0][j], v[1][j], v[2][j]);
  }

  // Exact 3x3 identity: med9 = med3(max of col-mins, med3 of col-meds,
  //                                 min of col-maxs).
  f4 res;
#pragma unroll
  for (int w = 0; w < 4; ++w) {
    const float a = max3(lo[w], lo[w + 1], lo[w + 2]);
    const float b = med3(mi[w], mi[w + 1], mi[w + 2]);
    const float c = min3(hi[w], hi[w + 1], hi[w + 2]);
    res[w] = med3(a, b, c);
  }

  const int gy = tileY + oy;
  const int gx = tileX + 4 * q;
  *(f4*)&dst[(size_t)gy * W + gx] = res;   // global_store_b128, coalesced
}

extern "C" void kernel_launch(void* const* d_in, const int* in_sizes, int n_in,
                              void* d_out, int out_size, void* d_ws,
                              size_t ws_size, hipStream_t stream) {
  (void)in_sizes; (void)n_in; (void)d_ws; (void)ws_size; (void)out_size;
  const float* x   = (const float*)d_in[0];
  float*       out = (float*)d_out;

  const int B = 128, H = 512, W = 512;
  dim3 grid(W / TW, H / TH, B);  // (4, 64, 128)
  median3x3_kernel<<<grid, NTHREADS, 0, stream>>>(x, out, H, W);
}